// SEMGFingerPredictor_7327214207621
// MI455X (gfx1250) — compile-verified
//
#include <hip/hip_runtime.h>
#include <math.h>

// Problem constants (match reference)
#define BATCH 256
#define NNODE 1024
#define DIN   3
#define HID   256
#define NOUT  5

#define ROWS  16          // batch rows per workgroup
#define LDSTR 260         // padded LDS activation stride (floats): conflict-free A reads
#define KC    32          // K-rows per TDM-staged weight slab (2 slabs double-buffered)

typedef __attribute__((ext_vector_type(2))) float v2f;
typedef __attribute__((ext_vector_type(8))) float v8f;
typedef __attribute__((ext_vector_type(4))) unsigned int u32x4;
typedef __attribute__((ext_vector_type(4))) int i32x4;
typedef __attribute__((ext_vector_type(8))) int i32x8;

// D = A(16x4,f32) * B(4x16,f32) + C(16x16,f32) on the CDNA5 matrix pipe.
__device__ __forceinline__ v8f wmma4(float a0, float a1, float b0, float b1, v8f c) {
  v2f A; A.x = a0; A.y = a1;
  v2f B; B.x = b0; B.y = b1;
  return __builtin_amdgcn_wmma_f32_16x16x4_f32(false, A, false, B, (short)0, c, false, false);
}

// Issue one TDM transfer: KC x 256 f32 tile (row-major, contiguous) global -> LDS.
// D# group0: count=1 | lds_addr | global_addr | type=2(image)
// D# group1: data_size=4B, tensor_dim0=256, tensor_dim1=KC, tile_dim0=256,
//            tile_dim1=KC, tensor_dim0_stride=256. Groups 2/3 unused (2D tensor).
__device__ __forceinline__ void tdm_load_slab(const float* gsrc, void* lds_dst) {
  const unsigned long long ga = (unsigned long long)(uintptr_t)gsrc;
  const unsigned int lds_off = (unsigned int)(uintptr_t)lds_dst; // low 32 bits = LDS byte offset

  u32x4 g0;
  g0[0] = 1u;                                            // count=1, user descriptor
  g0[1] = lds_off;                                       // lds_addr [63:32]
  g0[2] = (unsigned int)(ga & 0xFFFFFFFFu);              // global_addr [95:64]
  g0[3] = (unsigned int)((ga >> 32) & 0x01FFFFFFu)       // global_addr [120:96]
        | (2u << 30);                                    // type=2 [127:126]

  i32x8 g1;
  g1[0] = (int)(2u << 16);                               // data_size=2 (4 bytes); mask/flags=0
  g1[1] = (int)(256u << 16);                             // tensor_dim0[15:0]=256 at [63:48]
  g1[2] = (int)((unsigned)KC << 16);                     // dim0 hi=0 | tensor_dim1[15:0]=KC
  g1[3] = (int)(256u << 16);                             // dim1 hi=0 | tile_dim0=256 [127:112]
  g1[4] = (int)KC;                                       // tile_dim1=KC; tile_dim2=0
  g1[5] = 256;                                           // tensor_dim0_stride lo
  g1[6] = 0;                                             // stride0 hi | stride1 lo
  g1[7] = 0;                                             // stride1 hi

  i32x4 gz = {0, 0, 0, 0};
#if defined(__clang_major__) && (__clang_major__ >= 23)
  i32x8 gz8 = {0, 0, 0, 0, 0, 0, 0, 0};
  __builtin_amdgcn_tensor_load_to_lds(g0, g1, gz, gz, gz8, 0);
#else
  __builtin_amdgcn_tensor_load_to_lds(g0, g1, gz, gz, 0);
#endif
}

// One 16x256 = [16 x K=256] x [256 x 256] GEMM layer, TDM double-buffered.
// src/dst are padded LDS activation tiles; W row-major in global; bias length 256.
template <bool RELU>
__device__ __forceinline__ void gemm_layer(
    const float (&src)[ROWS][LDSTR], float (&dst)[ROWS][LDSTR],
    float (&slab)[2][KC][HID],
    const float* __restrict__ W, const float* __restrict__ bias,
    int wave, int l15, int half)
{
  const int col0 = (wave * 2 + 0) * 16 + l15;
  const int col1 = (wave * 2 + 1) * 16 + l15;
  v8f acc0 = {};
  v8f acc1 = {};

  if (wave == 0) tdm_load_slab(W, &slab[0][0][0]);       // prologue: chunk 0 in flight

  const int NCHUNK = HID / KC;                           // 8
  for (int ch = 0; ch < NCHUNK; ++ch) {
    if (wave == 0) {
      if (ch + 1 < NCHUNK) {
        tdm_load_slab(W + (size_t)(ch + 1) * KC * HID, &slab[(ch + 1) & 1][0][0]);
        __builtin_amdgcn_s_wait_tensorcnt(1);            // in-order: chunk ch resident
      } else {
        __builtin_amdgcn_s_wait_tensorcnt(0);
      }
    }
    __syncthreads();                                     // slab ch visible to all waves

    const float(*wb)[HID] = slab[ch & 1];
    #pragma unroll
    for (int kl = 0; kl < KC; kl += 4) {                 // 8 chained WMMAs / wave / chunk
      const int kk = kl + 2 * half;                      // per-lane K (fragment layout)
      const int kg = ch * KC + kk;
      const float a0 = src[l15][kg];
      const float a1 = src[l15][kg + 1];
      acc0 = wmma4(a0, a1, wb[kk][col0], wb[kk + 1][col0], acc0);
      acc1 = wmma4(a0, a1, wb[kk][col1], wb[kk + 1][col1], acc1);
    }
    __syncthreads();                                     // done reading before slab reuse
  }

  const float bias0 = bias[col0];
  const float bias1 = bias[col1];
  #pragma unroll
  for (int i = 0; i < 8; ++i) {                          // C/D: VGPR i -> rows i, i+8
    float v0 = acc0[i] + bias0;
    float v1 = acc1[i] + bias1;
    if (RELU) { v0 = fmaxf(v0, 0.0f); v1 = fmaxf(v1, 0.0f); }
    dst[i + 8 * half][col0] = v0;
    dst[i + 8 * half][col1] = v1;
  }
}

__global__ __launch_bounds__(256) void gcn_fused_kernel(
    const float* __restrict__ x,
    const float* __restrict__ W1, const float* __restrict__ b1,
    const float* __restrict__ W2, const float* __restrict__ b2,
    const float* __restrict__ Wr, const float* __restrict__ br,
    const float* __restrict__ Wf, const float* __restrict__ bf,
    float* __restrict__ out)
{
  __shared__ float sX[ROWS][4];            // mean(x), K padded 3 -> 4
  __shared__ float sH[ROWS][LDSTR];        // activation ping
  __shared__ float sG[ROWS][LDSTR];        // activation pong
  __shared__ float sW[2][KC][HID];         // TDM double-buffered weight slabs (64 KB)

  const int tid  = threadIdx.x;
  const int lane = tid & 31;
  const int wave = tid >> 5;
  const int rowBase = blockIdx.x * ROWS;

  const int half = lane >> 4;              // 0: lanes 0-15, 1: lanes 16-31
  const int l15  = lane & 15;

  // -------- Phase 0: xbar[b,:] = mean over N nodes (only HBM-heavy stage) ----
  for (int rr = 0; rr < 2; ++rr) {
    const int r = wave * 2 + rr;
    const float* xp = x + ((size_t)(rowBase + r) * NNODE) * DIN;
    float s0 = 0.f, s1 = 0.f, s2 = 0.f;
    for (int n = lane; n < NNODE; n += 32) {             // 12B/lane, coalesced
      const float* p = xp + n * 3;
      s0 += p[0]; s1 += p[1]; s2 += p[2];
    }
    for (int m = 16; m >= 1; m >>= 1) {                  // wave32 butterfly reduce
      s0 += __shfl_xor(s0, m, 32);
      s1 += __shfl_xor(s1, m, 32);
      s2 += __shfl_xor(s2, m, 32);
    }
    if (lane == 0) {
      sX[r][0] = s0 * (1.0f / NNODE);
      sX[r][1] = s1 * (1.0f / NNODE);
      sX[r][2] = s2 * (1.0f / NNODE);
      sX[r][3] = 0.0f;                                   // zero pad row K=3
    }
  }
  __syncthreads();

  // -------- Phase 1: sH = relu(xbar @ W1 + b1); single K=4 WMMA per tile -----
  {
    const float mk = (half == 0) ? 1.0f : 0.0f;          // branchless pad-row mask
    #pragma unroll
    for (int t = 0; t < 2; ++t) {
      const int col = (wave * 2 + t) * 16 + l15;
      const float a0 = sX[l15][2 * half + 0];
      const float a1 = sX[l15][2 * half + 1];
      const float b0  = W1[(2 * half) * HID + col];      // rows 0 / 2
      const float b1v = W1[1 * HID + col] * mk;          // row 1, or zero pad row 3
      v8f c = {};
      c = wmma4(a0, a1, b0, b1v, c);
      const float bias = b1[col];
      #pragma unroll
      for (int i = 0; i < 8; ++i) {
        const float v = c[i] + bias;
        sH[i + 8 * half][col] = v > 0.0f ? v : 0.0f;
      }
    }
  }
  __syncthreads();

  // -------- Phase 2: sG = sH @ W2 + b2 (TDM-staged, double-buffered) ---------
  gemm_layer<false>(sH, sG, sW, W2, b2, wave, l15, half);
  __syncthreads();

  // -------- Phase 3: sH = relu(sG @ Wr + br) ---------------------------------
  gemm_layer<true>(sG, sH, sW, Wr, br, wave, l15, half);
  __syncthreads();

  // -------- Phase 4: out = sigmoid(sH @ Wf + bf), OUT=5 padded to 16 ---------
  if (wave == 0) {                                       // uniform per-wave branch
    const float cmask = (l15 < NOUT) ? 1.0f : 0.0f;
    const int   ccol  = (l15 < NOUT) ? l15 : 0;          // clamped + masked
    v8f c = {};
    for (int k = 0; k < HID; k += 4) {
      const int kk = k + 2 * half;
      const float a0  = sH[l15][kk];
      const float a1  = sH[l15][kk + 1];
      const float b0  = Wf[(size_t)kk * NOUT + ccol] * cmask;
      const float b1v = Wf[(size_t)(kk + 1) * NOUT + ccol] * cmask;
      c = wmma4(a0, a1, b0, b1v, c);
    }
    const float bias = (l15 < NOUT) ? bf[l15] : 0.0f;
    #pragma unroll
    for (int i = 0; i < 8; ++i) {
      const int row = rowBase + i + 8 * half;
      const float z = c[i] + bias;
      const float s = 1.0f / (1.0f + __expf(-z));
      if (l15 < NOUT) out[row * NOUT + l15] = s;
    }
  }
}

extern "C" void kernel_launch(void* const* d_in, const int* in_sizes, int n_in,
                              void* d_out, int out_size, void* d_ws, size_t ws_size,
                              hipStream_t stream) {
  const float* x  = (const float*)d_in[0];
  const float* W1 = (const float*)d_in[1];
  const float* b1 = (const float*)d_in[2];
  const float* W2 = (const float*)d_in[3];
  const float* b2 = (const float*)d_in[4];
  const float* Wr = (const float*)d_in[5];
  const float* br = (const float*)d_in[6];
  const float* Wf = (const float*)d_in[7];
  const float* bf = (const float*)d_in[8];
  float* out = (float*)d_out;

  gcn_fused_kernel<<<BATCH / ROWS, 256, 0, stream>>>(x, W1, b1, W2, b2, Wr, br, Wf, bf, out);
}